// FNO1DDecoder_17617955848325
// MI455X (gfx1250) — compile-verified
//
#include <hip/hip_runtime.h>
#include <math.h>

// ---------------------------------------------------------------------------
// FNO1D decoder, fully fused for MI455X (gfx1250, wave32, WMMA f32 16x16x4).
//   modes[b][c][k] = (token @ w_dec + b_dec)   (k = interleaved re/im, 32)
//   x[b][c][t]     = sum_k modes * basis[k][t] (irfft == 32-wide GEMM)
//   out[b][t]      = gelu(x^T @ w1 + b1) @ w2 + b2   (t < 8190)
// ---------------------------------------------------------------------------

typedef __attribute__((ext_vector_type(2))) float v2f;
typedef __attribute__((ext_vector_type(8))) float v8f;

#define EMB     1024
#define NDEC    2048      // WIDTH * MODES1 * 2
#define WIDTHC  64
#define MODES   16
#define KDIM    32        // 2 * MODES (re/im interleaved)
#define LFULL   8192
#define LOUT    8190      // LFULL - PADDING
#define HID     128       // fc1 width

static __device__ __forceinline__ v8f wmma_f32(v2f a, v2f b, v8f c) {
    // D = A(16x4) * B(4x16) + C(16x16), fp32 matrix pipe
    return __builtin_amdgcn_wmma_f32_16x16x4_f32(
        /*neg_a=*/false, a, /*neg_b=*/false, b,
        /*c_mod=*/(short)0, c, /*reuse_a=*/false, /*reuse_b=*/false);
}

// ---------------------------------------------------------------------------
// Kernel 1: modes[64][2048] = token[64][1024] @ w_dec[1024][2048] + b_dec
// One wave per 16x16 tile; K-loop of 256 f32 WMMAs.
// grid (32, 4), block 128 (4 waves)
// ---------------------------------------------------------------------------
__global__ __launch_bounds__(128) void decode_gemm(
        const float* __restrict__ token, const float* __restrict__ w_dec,
        const float* __restrict__ b_dec, float* __restrict__ modes) {
    const int lane = threadIdx.x & 31;
    const int wave = threadIdx.x >> 5;
    const int lm = lane & 15;        // column within half (N or M index)
    const int hi = lane >> 4;        // half: K=0,1 vs K=2,3
    const int m0 = blockIdx.y * 16;
    const int n0 = (blockIdx.x * 4 + wave) * 16;

    v8f acc;
#pragma unroll
    for (int r = 0; r < 8; ++r) acc[r] = 0.0f;

    const float* arow = token + (m0 + lm) * EMB + 2 * hi;      // A: M=lm, K=2hi+j
    const float* bcol = w_dec + (2 * hi) * NDEC + n0 + lm;     // B: K=2hi+j, N=lm

    for (int k = 0; k < EMB; k += 4) {
        v2f a = *(const v2f*)(arow + k);
        v2f b;
        b.x = bcol[(k + 0) * NDEC];
        b.y = bcol[(k + 1) * NDEC];
        acc = wmma_f32(a, b, acc);
    }

    const float bias = b_dec[n0 + lm];
#pragma unroll
    for (int r = 0; r < 8; ++r)
        modes[(m0 + r + 8 * hi) * NDEC + n0 + lm] = acc[r] + bias;
}

// ---------------------------------------------------------------------------
// Kernel 2: basis[32][8192] (batch-independent irfft synthesis matrix; 1 MB,
// L2-resident, reused by all 64 batches)
// ---------------------------------------------------------------------------
__global__ void build_basis(float* __restrict__ basis) {
    const int t = blockIdx.x * blockDim.x + threadIdx.x;
    if (t >= LFULL) return;
    const float theta = 6.283185307179586f * (float)t / (float)LFULL;
    const float inv = 1.0f / (float)LFULL;
#pragma unroll
    for (int m = 0; m < MODES; ++m) {
        float s, c;
        __sincosf((float)m * theta, &s, &c);
        basis[(2 * m) * LFULL + t]     = ((m == 0) ? inv : 2.0f * inv) * c;
        basis[(2 * m + 1) * LFULL + t] = (m == 0) ? 0.0f : -2.0f * inv * s;
        // irfft ignores the imaginary part of the DC bin -> row 1 is zero
    }
}

// ---------------------------------------------------------------------------
// Kernel 3: fused spectral synthesis + MLP head.
// Block = 128 threads (4 waves); each block: one batch b, 64 time steps.
// Each wave owns 16 time steps:
//   stage B: x(64c x 16t) = modes(64x32) @ basis(32x16)   -> 32 WMMAs
//   stage C: h(16t x 128) = x^T(16x64)  @ w1(64x128)      -> 128 WMMAs
//   stage D: out[t] = sum_j gelu(h+b1)[t][j]*w2[j] + b2   -> shfl reduce
// grid (128, 64), block 128
// ---------------------------------------------------------------------------
__global__ __launch_bounds__(128) void fno_fused(
        const float* __restrict__ modes,  // [64][2048]
        const float* __restrict__ basis,  // [32][8192]
        const float* __restrict__ w1,     // [64][128]
        const float* __restrict__ b1,     // [128]
        const float* __restrict__ w2,     // [128]
        const float* __restrict__ b2,     // [1]
        float* __restrict__ out) {        // [64][8190]
    __shared__ float s_modes[WIDTHC * KDIM];   //  8 KB
    __shared__ float s_w1[WIDTHC * HID];       // 32 KB
    __shared__ float s_b1[HID];
    __shared__ float s_w2[HID];
    __shared__ float s_xT[4][16 * WIDTHC];     // 16 KB (per-wave transpose tile)

    const int b    = blockIdx.y;
    const int tid  = threadIdx.x;
    const int lane = tid & 31;
    const int wave = tid >> 5;
    const int lm   = lane & 15;
    const int hi   = lane >> 4;

    for (int i = tid; i < WIDTHC * KDIM; i += 128) s_modes[i] = modes[b * NDEC + i];
    for (int i = tid; i < WIDTHC * HID;  i += 128) s_w1[i] = w1[i];
    if (tid < HID) { s_b1[tid] = b1[tid]; s_w2[tid] = w2[tid]; }
    __syncthreads();

    const int t0 = (blockIdx.x * 4 + wave) * 16;

    // ---- stage B: spectral synthesis, x[c][t] for 16 t's ----
    v8f xacc[4];
#pragma unroll
    for (int cc = 0; cc < 4; ++cc)
#pragma unroll
        for (int r = 0; r < 8; ++r) xacc[cc][r] = 0.0f;

    for (int k = 0; k < KDIM; k += 4) {
        v2f bb;                                           // B: K=k+2hi+j, N=t
        bb.x = basis[(k + 2 * hi) * LFULL + t0 + lm];
        bb.y = basis[(k + 2 * hi + 1) * LFULL + t0 + lm];
#pragma unroll
        for (int cc = 0; cc < 4; ++cc) {                  // A: M=c, K=k+2hi+j
            v2f aa = *(const v2f*)(&s_modes[(cc * 16 + lm) * KDIM + k + 2 * hi]);
            xacc[cc] = wmma_f32(aa, bb, xacc[cc]);
        }
    }

    // transpose x (c-major D layout) -> xT[t][c] in this wave's LDS tile
    float* xT = &s_xT[wave][0];
#pragma unroll
    for (int cc = 0; cc < 4; ++cc)
#pragma unroll
        for (int r = 0; r < 8; ++r)
            xT[lm * WIDTHC + cc * 16 + hi * 8 + r] = xacc[cc][r];
    __syncthreads();   // order LDS write->read (cross-lane within wave)

    // ---- stage C: h = xT(16x64) @ w1(64x128) ----
    v8f hacc[8];
#pragma unroll
    for (int nt = 0; nt < 8; ++nt)
#pragma unroll
        for (int r = 0; r < 8; ++r) hacc[nt][r] = 0.0f;

    for (int kk = 0; kk < 16; ++kk) {                     // K = 4*kk .. 4*kk+3
        v2f aa = *(const v2f*)(&xT[lm * WIDTHC + kk * 4 + 2 * hi]);
#pragma unroll
        for (int nt = 0; nt < 8; ++nt) {
            v2f bb;
            bb.x = s_w1[(kk * 4 + 2 * hi) * HID + nt * 16 + lm];
            bb.y = s_w1[(kk * 4 + 2 * hi + 1) * HID + nt * 16 + lm];
            hacc[nt] = wmma_f32(aa, bb, hacc[nt]);
        }
    }

    // ---- stage D: exact gelu, dot with w2, 16-lane reduction ----
    float b1v[8], w2v[8];
#pragma unroll
    for (int nt = 0; nt < 8; ++nt) {
        b1v[nt] = s_b1[nt * 16 + lm];
        w2v[nt] = s_w2[nt * 16 + lm];
    }
    const float bias2 = b2[0];

#pragma unroll
    for (int r = 0; r < 8; ++r) {                         // t = t0 + r + 8*hi
        float p = 0.0f;
#pragma unroll
        for (int nt = 0; nt < 8; ++nt) {
            float v = hacc[nt][r] + b1v[nt];
            float g = 0.5f * v * (1.0f + erff(v * 0.7071067811865476f));
            p += g * w2v[nt];
        }
        // sum across the 16 lanes of this half (xor 1,2,4,8 stays in-half)
        p += __shfl_xor(p, 1);
        p += __shfl_xor(p, 2);
        p += __shfl_xor(p, 4);
        p += __shfl_xor(p, 8);
        if (lm == 0) {
            const int t = t0 + r + 8 * hi;
            if (t < LOUT) out[b * LOUT + t] = p + bias2;
        }
    }
}

// ---------------------------------------------------------------------------
extern "C" void kernel_launch(void* const* d_in, const int* in_sizes, int n_in,
                              void* d_out, int out_size, void* d_ws, size_t ws_size,
                              hipStream_t stream) {
    const float* token = (const float*)d_in[0];
    // d_in[1] = x_len scalar (compile-time 8192 here)
    const float* w_dec = (const float*)d_in[2];
    const float* b_dec = (const float*)d_in[3];
    const float* w1    = (const float*)d_in[4];
    const float* b1    = (const float*)d_in[5];
    const float* w2    = (const float*)d_in[6];
    const float* b2    = (const float*)d_in[7];
    float* out = (float*)d_out;

    float* modes = (float*)d_ws;            // 64 * 2048 f32 = 512 KB
    float* basis = modes + 64 * NDEC;       // 32 * 8192 f32 = 1 MB

    decode_gemm<<<dim3(32, 4), 128, 0, stream>>>(token, w_dec, b_dec, modes);
    build_basis<<<dim3(32), 256, 0, stream>>>(basis);
    fno_fused<<<dim3(128, 64), 128, 0, stream>>>(modes, basis, w1, b1, w2, b2, out);
    (void)in_sizes; (void)n_in; (void)out_size; (void)ws_size;
}